// GlobalSumPooling_45810121179453
// MI455X (gfx1250) — compile-verified
//
#include <hip/hip_runtime.h>

// out[b,d] = sum_t x[b,t,d];  x : f32[16, 4096, 1024] -> out : f32[16, 1024]
// HBM-bound (256 MiB @ 23.3 TB/s ~= 11.5 us floor). Reduction runs on the
// CDNA5 matrix pipe: D = ones(16x4) x B(4x16) + C accumulated over time chunks
// (v_wmma_f32_16x16x4_f32). Each wave owns TWO adjacent 16-wide column tiles
// (128 B contiguous per row) so every HBM line is fully consumed by one wave.

typedef __attribute__((ext_vector_type(2))) float v2f;
typedef __attribute__((ext_vector_type(8))) float v8f;

#define BATCH   16
#define TLEN    4096
#define DLEN    1024
#define CTILE   32             // columns per wave = 2 WMMA tiles = 128 B/row
#define WAVES   8              // waves per block (wave32 -> 256 threads)
#define TSLICE  (TLEN / WAVES) // 512 timesteps per wave

__global__ __launch_bounds__(256)
void gsp_wmma_kernel(const float* __restrict__ x, float* __restrict__ out) {
    __shared__ float partial[WAVES][CTILE];

    const int blk   = blockIdx.x;        // 0 .. 16*32-1
    const int b     = blk >> 5;          // batch
    const int ctile = blk & 31;          // which 32-wide column pair
    const int dbase = ctile * CTILE;

    const int lane = threadIdx.x & 31;
    const int wave = threadIdx.x >> 5;

    // B-matrix staging: lane supplies column n = lane&15; (VGPR, lane-half)
    // selects K. Lanes 0-15 load rows t0+{0,1}, lanes 16-31 rows t0+{2,3}.
    // A == ones and D is M-replicated, so any K permutation is sum-invariant.
    const int n    = lane & 15;
    const int koff = (lane >> 4) << 1;   // 0 or 2
    const int t0   = wave * TSLICE;

    const float* pA = x + ((size_t)b * TLEN + (size_t)(t0 + koff)) * DLEN + dbase + n;
    const float* pB = pA + 16;           // adjacent column tile (same rows)

    v2f ones; ones[0] = 1.0f; ones[1] = 1.0f;
    v8f cA0 = {}, cA1 = {}, cB0 = {}, cB1 = {};

    // 512 timesteps / 8 per iteration = 64 iterations, 4 WMMAs each.
    // Tile-A and tile-B loads for the same rows sit back-to-back in one
    // clause -> the 128-byte line of each row is consumed immediately.
    #pragma unroll 4
    for (int it = 0; it < TSLICE / 8; ++it) {
        v2f a0, a1, b0, b1;
        a0[0] = __builtin_nontemporal_load(pA);            // rows t+0 / t+2
        b0[0] = __builtin_nontemporal_load(pB);
        a0[1] = __builtin_nontemporal_load(pA + DLEN);     // rows t+1 / t+3
        b0[1] = __builtin_nontemporal_load(pB + DLEN);
        a1[0] = __builtin_nontemporal_load(pA + 4 * DLEN); // rows t+4 / t+6
        b1[0] = __builtin_nontemporal_load(pB + 4 * DLEN);
        a1[1] = __builtin_nontemporal_load(pA + 5 * DLEN); // rows t+5 / t+7
        b1[1] = __builtin_nontemporal_load(pB + 5 * DLEN);
        pA += 8 * DLEN;
        pB += 8 * DLEN;

        // (neg_a, A, neg_b, B, c_mod, C, reuse_a, reuse_b)
        cA0 = __builtin_amdgcn_wmma_f32_16x16x4_f32(false, ones, false, a0,
                                                    (short)0, cA0, false, false);
        cB0 = __builtin_amdgcn_wmma_f32_16x16x4_f32(false, ones, false, b0,
                                                    (short)0, cB0, false, false);
        cA1 = __builtin_amdgcn_wmma_f32_16x16x4_f32(false, ones, false, a1,
                                                    (short)0, cA1, false, false);
        cB1 = __builtin_amdgcn_wmma_f32_16x16x4_f32(false, ones, false, b1,
                                                    (short)0, cB1, false, false);
    }

    v8f cA = cA0 + cA1;                  // matrix add of partial sums
    v8f cB = cB0 + cB1;

    // D[m,n] replicated over m; lanes 0-15 hold their column's sum in c[0].
    if (lane < 16) {
        partial[wave][lane]      = cA[0];
        partial[wave][16 + lane] = cB[0];
    }
    __syncthreads();

    // Deterministic fixed-order combine of the 8 per-wave partials.
    if (threadIdx.x < CTILE) {
        float s = 0.0f;
        #pragma unroll
        for (int w = 0; w < WAVES; ++w) s += partial[w][threadIdx.x];
        out[(size_t)b * DLEN + dbase + threadIdx.x] = s;
    }
}

extern "C" void kernel_launch(void* const* d_in, const int* in_sizes, int n_in,
                              void* d_out, int out_size, void* d_ws, size_t ws_size,
                              hipStream_t stream) {
    (void)in_sizes; (void)n_in; (void)out_size; (void)d_ws; (void)ws_size;
    const float* x = (const float*)d_in[0];
    float* out     = (float*)d_out;

    dim3 grid(BATCH * (DLEN / CTILE));   // 512 blocks
    dim3 block(256);                      // 8 wave32 waves
    gsp_wmma_kernel<<<grid, block, 0, stream>>>(x, out);
}